// GraphConvolution_14078902797020
// MI455X (gfx1250) — compile-verified
//
#include <hip/hip_runtime.h>

// ---------------------------------------------------------------------------
// GraphConv: agg = segment_sum(h[src] * e_w, dst);  out = agg @ W + bias
// N=50000, E=800000, DIN=DOUT=256, all fp32.
//
// MI455X reasoning: edge gather/scatter (~0.9 GB, L2-resident at 192MB)
// dominates; GEMM is 6.55 GFLOP -> keep fp32 and use V_WMMA_F32_16X16X4_F32
// so the matrix phase runs on the WMMA pipe at full precision.
// ---------------------------------------------------------------------------

typedef float v2f __attribute__((ext_vector_type(2)));
typedef float v8f __attribute__((ext_vector_type(8)));

#define GC_N    50000
#define GC_E    800000
#define GC_DIM  256

// ---- kernel 1: zero the aggregation buffer (float4-vectorized stores) -----
__global__ void gc_zero_agg(float4* __restrict__ agg4, int n4) {
    int i = blockIdx.x * blockDim.x + threadIdx.x;
    if (i < n4) {
        agg4[i] = make_float4(0.f, 0.f, 0.f, 0.f);
    }
}

// ---- kernel 2: edge scatter: agg[dst[e]] += h[src[e]] * e_w[e] ------------
// 64 threads per edge -> edge index is wave-uniform (lanes of a wave share
// one edge), so src/dst/e_w loads scalarize; each lane handles 4 features
// with a float4 gather from h and 4 global f32 atomics into agg (L2 hit).
__global__ void gc_edge_scatter(const float*  __restrict__ h,
                                const float*  __restrict__ ew,
                                const int*    __restrict__ src,
                                const int*    __restrict__ dst,
                                float*        __restrict__ agg) {
    long long t = (long long)blockIdx.x * blockDim.x + threadIdx.x;
    int e = (int)(t >> 6);                 // 64 threads per edge
    if (e >= GC_E) return;
    int f = ((int)t & 63) << 2;            // feature chunk of 4

    int   s = src[e];                      // wave-uniform -> SGPR load
    int   d = dst[e];
    float w = ew[e];

    const float4 hv = *(const float4*)(h + (long long)s * GC_DIM + f);
    float* ap = agg + (long long)d * GC_DIM + f;

    atomicAdd(ap + 0, hv.x * w);
    atomicAdd(ap + 1, hv.y * w);
    atomicAdd(ap + 2, hv.z * w);
    atomicAdd(ap + 3, hv.w * w);
}

// ---- kernel 3: out = agg @ W + bias via V_WMMA_F32_16X16X4_F32 ------------
// One wave per 16x16 output tile. 50000 rows = 3125 M-tiles exactly; 16
// N-tiles. K=256 -> 64 WMMA(16x16x4) accumulations per tile.
//
// Fragment layouts (CDNA5 ISA 7.12.2, 32-bit, wave32):
//   A 16x4 : lane L<16 holds row M=L, K={k0,k0+1} in v0,v1;
//            lane L>=16 holds row M=L-16, K={k0+2,k0+3}.
//   B 4x16 : lane L<16 holds col N=L,  rows K={k0,k0+1} in v0,v1;
//            lane L>=16 holds col N=L-16, rows K={k0+2,k0+3}.
//   C/D    : 8 VGPRs; VGPR r, lane L -> M = r + 8*(L>=16), N = L&15.
__global__ void __launch_bounds__(256)
gc_gemm_wmma(const float* __restrict__ agg,
             const float* __restrict__ W,
             const float* __restrict__ bias,
             float*       __restrict__ out) {
    const int wave = threadIdx.x >> 5;
    const int lane = threadIdx.x & 31;
    const int tile = blockIdx.x * 8 + wave;     // 50000 tiles, grid = 6250*8
    const int tm = tile >> 4;                   // 0..3124
    const int tn = tile & 15;                   // 0..15
    const int m0 = tm << 4;
    const int n0 = tn << 4;

    const int half = lane >> 4;                 // 0: K pair {0,1}, 1: {2,3}
    const int l15  = lane & 15;

    const float* __restrict__ arow = agg + (long long)(m0 + l15) * GC_DIM;
    const float* __restrict__ bcol = W + n0 + l15;

    v8f c = {};
#pragma unroll 4
    for (int k0 = 0; k0 < GC_DIM; k0 += 4) {
        const int kk = k0 + (half << 1);
        v2f a, b;
        a.x = arow[kk];                          // contiguous -> b64 load
        a.y = arow[kk + 1];
        b.x = bcol[(long long)kk * GC_DIM];      // two rows of W
        b.y = bcol[(long long)(kk + 1) * GC_DIM];
        // 8 args: (neg_a, A, neg_b, B, c_mod, C, reuse_a, reuse_b)
        c = __builtin_amdgcn_wmma_f32_16x16x4_f32(
                false, a, false, b, (short)0, c, false, false);
    }

    const float bv = bias[n0 + l15];
    const int mbase = m0 + (half << 3);
#pragma unroll
    for (int r = 0; r < 8; ++r) {
        out[(long long)(mbase + r) * GC_DIM + n0 + l15] = c[r] + bv;
    }
}

// ---------------------------------------------------------------------------
extern "C" void kernel_launch(void* const* d_in, const int* in_sizes, int n_in,
                              void* d_out, int out_size, void* d_ws, size_t ws_size,
                              hipStream_t stream) {
    const float* h    = (const float*)d_in[0];   // [N, 256]
    const float* ew   = (const float*)d_in[1];   // [E, 1]
    const int*   src  = (const int*)d_in[2];     // [E]
    const int*   dst  = (const int*)d_in[3];     // [E]
    const float* W    = (const float*)d_in[4];   // [256, 256]
    const float* bias = (const float*)d_in[5];   // [256]
    float*       out  = (float*)d_out;           // [N, 256]
    float*       agg  = (float*)d_ws;            // [N, 256] scratch (51.2 MB)

    // 1) zero accumulator (must happen every call; ws is poisoned by harness)
    {
        const int n4 = GC_N * GC_DIM / 4;        // 3,200,000 float4
        const int blk = 256;
        gc_zero_agg<<<(n4 + blk - 1) / blk, blk, 0, stream>>>((float4*)agg, n4);
    }

    // 2) gather + scale + scatter-add
    {
        const long long work = (long long)GC_E * 64;     // 64 threads/edge
        const int blk = 256;
        const int grid = (int)((work + blk - 1) / blk);  // 200,000 blocks
        gc_edge_scatter<<<grid, blk, 0, stream>>>(h, ew, src, dst, agg);
    }

    // 3) WMMA GEMM + bias
    {
        const int tiles = (GC_N / 16) * (GC_DIM / 16);   // 50,000
        gc_gemm_wmma<<<tiles / 8, 256, 0, stream>>>(agg, W, bias, out);
    }
}